// CD_GCN_Net_21663815041318
// MI455X (gfx1250) — compile-verified
//
#include <hip/hip_runtime.h>
#include <hip/hip_bf16.h>

typedef __attribute__((ext_vector_type(2))) float v2f;
typedef __attribute__((ext_vector_type(8))) float v8f;

#define BN_EPS 1e-5f

// ---------------------------------------------------------------------------
// generic zero
// ---------------------------------------------------------------------------
__global__ void zero_k(float* __restrict__ p, size_t n) {
    size_t i = (size_t)blockIdx.x * blockDim.x + threadIdx.x;
    if (i < n) p[i] = 0.0f;
}

// ---------------------------------------------------------------------------
// agg init, layer 1:  agg[i] = {x0[i], 0, ..., 0}   (padded [N,16] row)
// ---------------------------------------------------------------------------
__global__ void init_agg_x0_k(const float* __restrict__ x0, float* __restrict__ agg, size_t n) {
    size_t i = (size_t)blockIdx.x * blockDim.x + threadIdx.x;
    if (i >= n) return;
    float4 z; z.x = 0.f; z.y = 0.f; z.z = 0.f; z.w = 0.f;
    float4 a0 = z; a0.x = x0[i];
    float4* p = (float4*)(agg + i * 16);
    p[0] = a0; p[1] = z; p[2] = z; p[3] = z;
}

// ---------------------------------------------------------------------------
// agg init, layers 2/3:  agg[i] = x_l[i]  (padded row copy, pads already 0)
// ---------------------------------------------------------------------------
__global__ void copy16_k(const float* __restrict__ xl, float* __restrict__ agg, size_t n) {
    size_t i = (size_t)blockIdx.x * blockDim.x + threadIdx.x;
    if (i >= n) return;
    const float4* s = (const float4*)(xl + i * 16);
    float4*       p = (float4*)(agg + i * 16);
    p[0] = s[0]; p[1] = s[1]; p[2] = s[2]; p[3] = s[3];
}

// ---------------------------------------------------------------------------
// pad weights/biases into wpad:
//   [0..255]   Wa padded 16x16   (row-major, zeros outside d_in x 10)
//   [256..511] Wb padded 16x16
//   [512..527] ba padded 16
//   [528..543] bb padded 16
// ---------------------------------------------------------------------------
__global__ void pad_weights_k(const float* __restrict__ Wa, int d_in,
                              const float* __restrict__ ba,
                              const float* __restrict__ Wb,
                              const float* __restrict__ bb,
                              float* __restrict__ wpad) {
    for (int i = threadIdx.x; i < 544; i += blockDim.x) {
        float v = 0.f;
        if (i < 256) {
            int r = i >> 4, c = i & 15;
            if (r < d_in && c < 10) v = Wa[r * 10 + c];
        } else if (i < 512) {
            int j = i - 256, r = j >> 4, c = j & 15;
            if (r < 10 && c < 10) v = Wb[r * 10 + c];
        } else if (i < 528) {
            int c = i - 512;
            if (c < 10) v = ba[c];
        } else {
            int c = i - 528;
            if (c < 10) v = bb[c];
        }
        wpad[i] = v;
    }
}

// ---------------------------------------------------------------------------
// edge scatter: agg[dst][k] += xin[src][k]   (thread per (edge,feature))
// ---------------------------------------------------------------------------
template <int D, int XS>
__global__ void scatter_add_k(const int* __restrict__ srcI, const int* __restrict__ dstI,
                              const float* __restrict__ xin, float* __restrict__ agg,
                              unsigned int total) {
    unsigned int gid = blockIdx.x * blockDim.x + threadIdx.x;
    if (gid >= total) return;
    unsigned int e = gid / D;           // compile-time divisor -> mul/shift
    unsigned int k = gid - e * D;
    int s = srcI[e];
    int t = dstI[e];
    atomicAdd(&agg[(size_t)t * 16 + k], xin[(size_t)s * XS + k]);
}

// ---------------------------------------------------------------------------
// GIN MLP via f32 WMMA.  One wave (32 threads) owns a 16-node tile.
//   H1 = agg @ Wa + ba     (agg pre-initialized to x + segment_sum, zero-padded)
//   H2 = H1  @ Wb + bb
// Branch-free: all pads are zero in agg/wpad, all loads unconditional.
// f32 WMMA fragment layout (ISA 7.12.2):
//   A 16x4 : lane L -> M = L&15, K pair base = 2*(L>>4); regs {K, K+1}
//   B 4x16 : lane L -> N = L&15, K pair base = 2*(L>>4); regs {K, K+1}
//   C/D    : lane L -> N = L&15, reg v -> M = v + 8*(L>>4)
// ---------------------------------------------------------------------------
template <int NKK>
__global__ __launch_bounds__(32)
void gin_mlp_wmma(const float* __restrict__ agg,
                  const float* __restrict__ wpad,
                  float* __restrict__ hout,
                  float* __restrict__ colsum, float* __restrict__ colsq) {
    __shared__ float lds[16 * 16];

    const int lane = threadIdx.x;       // 0..31 (wave32)
    const int hf   = lane >> 4;         // half: 0 or 1
    const int lp   = lane & 15;
    const size_t row0 = (size_t)blockIdx.x * 16;

    const float* Wa = wpad;
    const float* Wb = wpad + 256;
    const float biasA = wpad[512 + lp];
    const float biasB = wpad[528 + lp];

    // B fragments (unconditional, from padded weights)
    v2f bA[NKK], bB[3];
    #pragma unroll
    for (int kk = 0; kk < NKK; ++kk) {
        int k0 = 4 * kk + 2 * hf;
        bA[kk].x = Wa[k0 * 16 + lp];
        bA[kk].y = Wa[(k0 + 1) * 16 + lp];
    }
    #pragma unroll
    for (int kk = 0; kk < 3; ++kk) {
        int k0 = 4 * kk + 2 * hf;
        bB[kk].x = Wb[k0 * 16 + lp];
        bB[kk].y = Wb[(k0 + 1) * 16 + lp];
    }

    // ---- GEMM1: H1 = agg @ Wa + ba ----
    const float* arow = agg + (row0 + lp) * 16 + 2 * hf;   // 8B-aligned
    v8f acc = {};
    #pragma unroll
    for (int kk = 0; kk < NKK; ++kk) {
        v2f a = *(const v2f*)(arow + 4 * kk);              // global_load_b64
        acc = __builtin_amdgcn_wmma_f32_16x16x4_f32(false, a, false, bA[kk],
                                                    (short)0, acc, false, false);
    }
    #pragma unroll
    for (int v = 0; v < 8; ++v) acc[v] += biasA;

    // D-layout -> row-major LDS tile -> A-layout for GEMM2
    #pragma unroll
    for (int v = 0; v < 8; ++v) lds[(v + 8 * hf) * 16 + lp] = acc[v];
    __syncthreads();   // blockDim == 32: wave-level LDS ordering

    // ---- GEMM2: H2 = H1 @ Wb + bb ----
    const float* lrow = lds + lp * 16 + 2 * hf;
    v8f acc2 = {};
    #pragma unroll
    for (int kk = 0; kk < 3; ++kk) {
        v2f a = *(const v2f*)(lrow + 4 * kk);              // ds_load_b64
        acc2 = __builtin_amdgcn_wmma_f32_16x16x4_f32(false, a, false, bB[kk],
                                                     (short)0, acc2, false, false);
    }
    #pragma unroll
    for (int v = 0; v < 8; ++v) acc2[v] += biasB;

    // ---- store H2, accumulate BN stats ----
    float s1 = 0.f, s2 = 0.f;
    #pragma unroll
    for (int v = 0; v < 8; ++v) {
        float h = acc2[v];
        hout[(row0 + v + 8 * hf) * 16 + lp] = h;
        s1 += h;
        s2 += h * h;
    }
    // lanes L and L+16 hold the same column N=lp for different row halves
    s1 += __shfl_xor(s1, 16, 32);
    s2 += __shfl_xor(s2, 16, 32);
    if (hf == 0 && lp < 10) {
        atomicAdd(&colsum[lp], s1);
        atomicAdd(&colsq[lp], s2);
    }
}

// ---------------------------------------------------------------------------
// BN: fold (g, be, mean, var) into per-feature scale/shift
// ---------------------------------------------------------------------------
__global__ void bn_finalize_k(const float* __restrict__ colsum, const float* __restrict__ colsq,
                              const float* __restrict__ g, const float* __restrict__ be,
                              float* __restrict__ scale, float* __restrict__ shift, float invN) {
    int n = threadIdx.x;                // 0..15
    float sc = 0.f, sh = 0.f;
    if (n < 10) {
        float mu  = colsum[n] * invN;
        float var = colsq[n] * invN - mu * mu;
        float r   = rsqrtf(var + BN_EPS);
        sc = g[n] * r;
        sh = be[n] - sc * mu;
    }
    scale[n] = sc;
    shift[n] = sh;
}

// ---------------------------------------------------------------------------
// apply BN affine -> x_l  (padded [N,16], cols 10..15 = 0) and
// accumulate this layer's slice of concat @ Wf into logits [N,4]
// ---------------------------------------------------------------------------
__global__ void bn_apply_logits_k(const float* __restrict__ h,
                                  const float* __restrict__ scale,
                                  const float* __restrict__ shift,
                                  const float* __restrict__ Wf,   // already offset to layer rows
                                  const float* __restrict__ bf,   // non-null only for layer 1
                                  float* __restrict__ xl,
                                  float* __restrict__ logits, size_t n) {
    size_t i = (size_t)blockIdx.x * blockDim.x + threadIdx.x;
    if (i >= n) return;
    float lg0, lg1, lg2, lg3;
    if (bf) {
        lg0 = bf[0]; lg1 = bf[1]; lg2 = bf[2]; lg3 = bf[3];
    } else {
        float4 l4 = ((const float4*)logits)[i];
        lg0 = l4.x; lg1 = l4.y; lg2 = l4.z; lg3 = l4.w;
    }
    const float4* hr = (const float4*)(h + i * 16);
    float4 h0 = hr[0], h1 = hr[1], h2 = hr[2];
    float hv[12] = {h0.x, h0.y, h0.z, h0.w, h1.x, h1.y, h1.z, h1.w,
                    h2.x, h2.y, h2.z, h2.w};
    float xr[16];
    #pragma unroll
    for (int k = 0; k < 10; ++k) {
        float v = scale[k] * hv[k] + shift[k];
        xr[k] = v;
        lg0 += v * Wf[k * 4 + 0];
        lg1 += v * Wf[k * 4 + 1];
        lg2 += v * Wf[k * 4 + 2];
        lg3 += v * Wf[k * 4 + 3];
    }
    #pragma unroll
    for (int k = 10; k < 16; ++k) xr[k] = 0.f;

    float4* xo = (float4*)(xl + i * 16);
    #pragma unroll
    for (int j = 0; j < 4; ++j) {
        float4 t; t.x = xr[4*j]; t.y = xr[4*j+1]; t.z = xr[4*j+2]; t.w = xr[4*j+3];
        xo[j] = t;
    }
    float4 lo; lo.x = lg0; lo.y = lg1; lo.z = lg2; lo.w = lg3;
    ((float4*)logits)[i] = lo;
}

// ---------------------------------------------------------------------------
// softmax over 4 logits -> s (written straight to d_out)
// ---------------------------------------------------------------------------
__global__ void softmax_k(const float* __restrict__ logits, float* __restrict__ out, size_t n) {
    size_t i = (size_t)blockIdx.x * blockDim.x + threadIdx.x;
    if (i >= n) return;
    float4 l = ((const float4*)logits)[i];
    float m = fmaxf(fmaxf(l.x, l.y), fmaxf(l.z, l.w));
    float e0 = expf(l.x - m), e1 = expf(l.y - m), e2 = expf(l.z - m), e3 = expf(l.w - m);
    float inv = 1.f / (e0 + e1 + e2 + e3);
    float4 o; o.x = e0 * inv; o.y = e1 * inv; o.z = e2 * inv; o.w = e3 * inv;
    ((float4*)out)[i] = o;
}

// ---------------------------------------------------------------------------
// fused modularity edge pass: deg[src] += w;  acc[0] += w (2m);
// acc[1] += w * dot(s[src], s[dst])
// ---------------------------------------------------------------------------
__global__ void edge_mod_k(const int* __restrict__ srcI, const int* __restrict__ dstI,
                           const float* __restrict__ w, const float* __restrict__ s,
                           float* __restrict__ deg, float* __restrict__ acc, size_t E_) {
    size_t e = (size_t)blockIdx.x * blockDim.x + threadIdx.x;
    float we = 0.f, p = 0.f;
    if (e < E_) {
        int a = srcI[e];
        int b = dstI[e];
        we = w[e];
        atomicAdd(&deg[a], we);
        float4 sa = ((const float4*)s)[a];
        float4 sb = ((const float4*)s)[b];
        p = we * (sa.x * sb.x + sa.y * sb.y + sa.z * sb.z + sa.w * sb.w);
    }
    #pragma unroll
    for (int o = 16; o > 0; o >>= 1) {
        we += __shfl_down(we, o, 32);
        p  += __shfl_down(p,  o, 32);
    }
    if ((threadIdx.x & 31) == 0) {
        atomicAdd(&acc[0], we);
        atomicAdd(&acc[1], p);
    }
}

// ---------------------------------------------------------------------------
// ds[c] = sum_i deg[i] * s[i][c]
// ---------------------------------------------------------------------------
__global__ void ds_k(const float* __restrict__ deg, const float* __restrict__ s,
                     float* __restrict__ ds, size_t n) {
    size_t i = (size_t)blockIdx.x * blockDim.x + threadIdx.x;
    float p0 = 0.f, p1 = 0.f, p2 = 0.f, p3 = 0.f;
    if (i < n) {
        float d = deg[i];
        float4 sv = ((const float4*)s)[i];
        p0 = d * sv.x; p1 = d * sv.y; p2 = d * sv.z; p3 = d * sv.w;
    }
    #pragma unroll
    for (int o = 16; o > 0; o >>= 1) {
        p0 += __shfl_down(p0, o, 32);
        p1 += __shfl_down(p1, o, 32);
        p2 += __shfl_down(p2, o, 32);
        p3 += __shfl_down(p3, o, 32);
    }
    if ((threadIdx.x & 31) == 0) {
        atomicAdd(&ds[0], p0);
        atomicAdd(&ds[1], p1);
        atomicAdd(&ds[2], p2);
        atomicAdd(&ds[3], p3);
    }
}

// ---------------------------------------------------------------------------
// q = pos/2m - sum((ds/2m)^2)
// ---------------------------------------------------------------------------
__global__ void finalize_q_k(const float* __restrict__ acc, float* __restrict__ qout) {
    float inv = 1.f / acc[0];
    float d0 = acc[2] * inv, d1 = acc[3] * inv, d2 = acc[4] * inv, d3 = acc[5] * inv;
    *qout = acc[1] * inv - (d0 * d0 + d1 * d1 + d2 * d2 + d3 * d3);
}

// ---------------------------------------------------------------------------
extern "C" void kernel_launch(void* const* d_in, const int* in_sizes, int n_in,
                              void* d_out, int out_size, void* d_ws, size_t ws_size,
                              hipStream_t stream) {
    const float* x0   = (const float*)d_in[0];
    const int*   ei   = (const int*)d_in[1];
    const float* wat  = (const float*)d_in[2];

    const size_t Nn = (size_t)in_sizes[0];     // N (IN == 1)
    const size_t E_ = (size_t)in_sizes[2];
    const int* srcI = ei;
    const int* dstI = ei + E_;

    const float* W1a = (const float*)d_in[3];
    const float* b1a = (const float*)d_in[4];
    const float* W1b = (const float*)d_in[5];
    const float* b1b = (const float*)d_in[6];
    const float* g1  = (const float*)d_in[7];
    const float* be1 = (const float*)d_in[8];
    const float* W2a = (const float*)d_in[9];
    const float* b2a = (const float*)d_in[10];
    const float* W2b = (const float*)d_in[11];
    const float* b2b = (const float*)d_in[12];
    const float* g2  = (const float*)d_in[13];
    const float* be2 = (const float*)d_in[14];
    const float* W3a = (const float*)d_in[15];
    const float* b3a = (const float*)d_in[16];
    const float* W3b = (const float*)d_in[17];
    const float* b3b = (const float*)d_in[18];
    const float* g3  = (const float*)d_in[19];
    const float* be3 = (const float*)d_in[20];
    const float* Wf  = (const float*)d_in[21];
    const float* bf  = (const float*)d_in[22];

    float* out = (float*)d_out;

    // workspace layout (floats)
    float* ws = (float*)d_ws;
    const size_t NP = Nn * 16;
    float* x1     = ws;
    float* x2     = ws + NP;
    float* x3     = ws + 2 * NP;
    float* hb     = ws + 3 * NP;
    float* agg    = ws + 4 * NP;
    float* logits = ws + 5 * NP;           // [N,4]
    float* deg    = logits + Nn * 4;       // [N]
    float* stat   = deg + Nn;
    float* colsum = stat;                  // [16]
    float* colsq  = stat + 16;             // [16]
    float* scale  = stat + 32;             // [16]
    float* shift  = stat + 48;             // [16]
    float* acc    = stat + 64;             // [0]=2m [1]=pos [2..5]=ds
    float* wpad   = stat + 80;             // [544] padded weights/biases

    auto cdiv = [](size_t a, size_t b) { return (unsigned)((a + b - 1) / b); };
    const unsigned gN   = cdiv(Nn, 256);
    const unsigned gE   = cdiv(E_, 256);
    const unsigned gE10 = cdiv(E_ * 10, 256);
    const unsigned tiles = (unsigned)(Nn / 16);   // N % 16 == 0
    const float invN = 1.0f / (float)Nn;

    // ---- layer 1 (d_in = 1) ----
    init_agg_x0_k<<<gN, 256, 0, stream>>>(x0, agg, Nn);
    zero_k<<<1, 32, 0, stream>>>(colsum, 32);
    pad_weights_k<<<1, 256, 0, stream>>>(W1a, 1, b1a, W1b, b1b, wpad);
    scatter_add_k<1, 1><<<gE, 256, 0, stream>>>(srcI, dstI, x0, agg, (unsigned)E_);
    gin_mlp_wmma<1><<<tiles, 32, 0, stream>>>(agg, wpad, hb, colsum, colsq);
    bn_finalize_k<<<1, 16, 0, stream>>>(colsum, colsq, g1, be1, scale, shift, invN);
    bn_apply_logits_k<<<gN, 256, 0, stream>>>(hb, scale, shift, Wf + 0, bf, x1, logits, Nn);

    // ---- layer 2 (d_in = 10) ----
    copy16_k<<<gN, 256, 0, stream>>>(x1, agg, Nn);
    zero_k<<<1, 32, 0, stream>>>(colsum, 32);
    pad_weights_k<<<1, 256, 0, stream>>>(W2a, 10, b2a, W2b, b2b, wpad);
    scatter_add_k<10, 16><<<gE10, 256, 0, stream>>>(srcI, dstI, x1, agg, (unsigned)(E_ * 10));
    gin_mlp_wmma<3><<<tiles, 32, 0, stream>>>(agg, wpad, hb, colsum, colsq);
    bn_finalize_k<<<1, 16, 0, stream>>>(colsum, colsq, g2, be2, scale, shift, invN);
    bn_apply_logits_k<<<gN, 256, 0, stream>>>(hb, scale, shift, Wf + 40, nullptr, x2, logits, Nn);

    // ---- layer 3 (d_in = 10) ----
    copy16_k<<<gN, 256, 0, stream>>>(x2, agg, Nn);
    zero_k<<<1, 32, 0, stream>>>(colsum, 32);
    pad_weights_k<<<1, 256, 0, stream>>>(W3a, 10, b3a, W3b, b3b, wpad);
    scatter_add_k<10, 16><<<gE10, 256, 0, stream>>>(srcI, dstI, x2, agg, (unsigned)(E_ * 10));
    gin_mlp_wmma<3><<<tiles, 32, 0, stream>>>(agg, wpad, hb, colsum, colsq);
    bn_finalize_k<<<1, 16, 0, stream>>>(colsum, colsq, g3, be3, scale, shift, invN);
    bn_apply_logits_k<<<gN, 256, 0, stream>>>(hb, scale, shift, Wf + 80, nullptr, x3, logits, Nn);

    // ---- softmax + modularity ----
    zero_k<<<cdiv(Nn, 256), 256, 0, stream>>>(deg, Nn);
    zero_k<<<1, 32, 0, stream>>>(acc, 6);
    softmax_k<<<gN, 256, 0, stream>>>(logits, out, Nn);
    edge_mod_k<<<gE, 256, 0, stream>>>(srcI, dstI, wat, out, deg, acc, E_);
    ds_k<<<gN, 256, 0, stream>>>(deg, out, acc + 2, Nn);
    finalize_q_k<<<1, 1, 0, stream>>>(acc, out + Nn * 4);
}